// GAT_5239860101746
// MI455X (gfx1250) — compile-verified
//
#include <hip/hip_runtime.h>
#include <math.h>

#define Nn 4096
#define Ff 512
#define Hh 8
#define Dd 16
#define Cc 16
#define CAP 128
#define LRELU_ALPHA 0.2f

typedef __attribute__((ext_vector_type(2))) float v2f;
typedef __attribute__((ext_vector_type(8))) float v8f;

// ---------------------------------------------------------------------------
// CSR build: one block per row, b128 (int4) coalesced scan of the 64 MB
// adjacency matrix (the bandwidth-dominant step), compacting nonzero columns.
// ---------------------------------------------------------------------------
__global__ __launch_bounds__(256) void gat_csr(const int* __restrict__ adj,
                                               int* __restrict__ cnt,
                                               int* __restrict__ nbrs) {
  const int n = blockIdx.x;
  const int4* row4 = (const int4*)(adj + (size_t)n * Nn);
  // Nn/4 = 1024 int4 elements per row; 256 threads -> 4 iterations.
  for (int q = threadIdx.x; q < Nn / 4; q += blockDim.x) {
    const int4 v = row4[q];
    const int m0 = q * 4;
    if (v.x != 0) { int p = atomicAdd(&cnt[n], 1); if (p < CAP) nbrs[n * CAP + p] = m0 + 0; }
    if (v.y != 0) { int p = atomicAdd(&cnt[n], 1); if (p < CAP) nbrs[n * CAP + p] = m0 + 1; }
    if (v.z != 0) { int p = atomicAdd(&cnt[n], 1); if (p < CAP) nbrs[n * CAP + p] = m0 + 2; }
    if (v.w != 0) { int p = atomicAdd(&cnt[n], 1); if (p < CAP) nbrs[n * CAP + p] = m0 + 3; }
  }
}

// ---------------------------------------------------------------------------
// GEMM1: Wh1[h,n,d] = sum_f x[n,f] * W1[h,f,d].  One wave per (16-row tile, h).
// fp32 WMMA 16x16x4, K-loop over F=512.
// ---------------------------------------------------------------------------
__global__ __launch_bounds__(32) void gat_gemm1(const float* __restrict__ x,
                                                const float* __restrict__ W1,
                                                float* __restrict__ Wh1) {
  const int m0    = blockIdx.x * 16;
  const int h     = blockIdx.y;
  const int lane  = threadIdx.x;
  const int row   = lane & 15;   // M index (A) / N index (B,D)
  const int khalf = lane >> 4;   // 0: K pair {0,1}, 1: K pair {2,3}

  v8f c = {};
  const float* xb = x + (size_t)(m0 + row) * Ff;
  const float* wb = W1 + (size_t)h * Ff * Dd + row;  // column = row

  for (int k0 = 0; k0 < Ff; k0 += 4) {
    const int ka = k0 + 2 * khalf;
    v2f a, b;
    a.x = xb[ka];
    a.y = xb[ka + 1];
    b.x = wb[(size_t)ka * Dd];
    b.y = wb[(size_t)(ka + 1) * Dd];
    c = __builtin_amdgcn_wmma_f32_16x16x4_f32(false, a, false, b,
                                              (short)0, c, false, false);
  }
  // D layout: VGPR r -> lanes 0-15: M=r, N=lane; lanes 16-31: M=r+8, N=lane-16
  float* out = Wh1 + ((size_t)h * Nn + m0) * Dd + row;
#pragma unroll
  for (int r = 0; r < 8; ++r) out[(size_t)(r + 8 * khalf) * Dd] = c[r];
}

// ---------------------------------------------------------------------------
// GEMM2: Wh2[n,c] = sum_k hcat[n,k] * Wout[k,c].  K = H*D = 128.
// ---------------------------------------------------------------------------
__global__ __launch_bounds__(32) void gat_gemm2(const float* __restrict__ hcat,
                                                const float* __restrict__ Wout,
                                                float* __restrict__ Wh2) {
  const int m0    = blockIdx.x * 16;
  const int lane  = threadIdx.x;
  const int row   = lane & 15;
  const int khalf = lane >> 4;

  v8f c = {};
  const float* xb = hcat + (size_t)(m0 + row) * (Hh * Dd);
  const float* wb = Wout + row;

  for (int k0 = 0; k0 < Hh * Dd; k0 += 4) {
    const int ka = k0 + 2 * khalf;
    v2f a, b;
    a.x = xb[ka];
    a.y = xb[ka + 1];
    b.x = wb[(size_t)ka * Cc];
    b.y = wb[(size_t)(ka + 1) * Cc];
    c = __builtin_amdgcn_wmma_f32_16x16x4_f32(false, a, false, b,
                                              (short)0, c, false, false);
  }
  float* out = Wh2 + (size_t)m0 * Cc + row;
#pragma unroll
  for (int r = 0; r < 8; ++r) out[(size_t)(r + 8 * khalf) * Cc] = c[r];
}

// ---------------------------------------------------------------------------
// Attention score terms, layer 1: f1[h,n] = Wh1[h,n,:].a1[h,:D], f2 likewise.
// ---------------------------------------------------------------------------
__global__ __launch_bounds__(256) void gat_scores1(const float* __restrict__ Wh1,
                                                   const float* __restrict__ a1,
                                                   float* __restrict__ f1,
                                                   float* __restrict__ f2) {
  const int idx = blockIdx.x * blockDim.x + threadIdx.x;
  if (idx >= Hh * Nn) return;
  const int h = idx / Nn;
  const float* w  = Wh1 + (size_t)idx * Dd;
  const float* av = a1 + h * 2 * Dd;
  float s1 = 0.f, s2 = 0.f;
#pragma unroll
  for (int d = 0; d < Dd; ++d) {
    const float v = w[d];
    s1 += v * av[d];
    s2 += v * av[Dd + d];
  }
  f1[idx] = s1;
  f2[idx] = s2;
}

__global__ __launch_bounds__(256) void gat_scores2(const float* __restrict__ Wh2,
                                                   const float* __restrict__ a_out,
                                                   float* __restrict__ f1o,
                                                   float* __restrict__ f2o) {
  const int n = blockIdx.x * blockDim.x + threadIdx.x;
  if (n >= Nn) return;
  const float* w = Wh2 + (size_t)n * Cc;
  float s1 = 0.f, s2 = 0.f;
#pragma unroll
  for (int c = 0; c < Cc; ++c) {
    const float v = w[c];
    s1 += v * a_out[c];
    s2 += v * a_out[Cc + c];
  }
  f1o[n] = s1;
  f2o[n] = s2;
}

// ---------------------------------------------------------------------------
// Layer-1 sparse attention + aggregation + ELU + head concat.
// One block (8 waves) per node n; wave h handles head h.
// ---------------------------------------------------------------------------
__global__ __launch_bounds__(256) void gat_attn1(const int* __restrict__ cnt,
                                                 const int* __restrict__ nbrs,
                                                 const float* __restrict__ f1,
                                                 const float* __restrict__ f2,
                                                 const float* __restrict__ Wh1,
                                                 float* __restrict__ hcat) {
  __shared__ int   s_nbr[CAP];
  __shared__ float s_w[Hh * CAP];

  const int n   = blockIdx.x;
  const int tid = threadIdx.x;
  const int deg = min(cnt[n], CAP);
  if (tid < deg) s_nbr[tid] = nbrs[n * CAP + tid];
  __syncthreads();

  const int   h    = tid >> 5;
  const int   lane = tid & 31;
  const float f1h  = f1[h * Nn + n];

  // pass 1: row max of leaky_relu(f1 + f2)
  float mx = -3.4e38f;
  for (int i = lane; i < deg; i += 32) {
    float e = f1h + f2[h * Nn + s_nbr[i]];
    e = (e >= 0.f) ? e : LRELU_ALPHA * e;
    mx = fmaxf(mx, e);
  }
#pragma unroll
  for (int o = 16; o > 0; o >>= 1) mx = fmaxf(mx, __shfl_xor(mx, o, 32));

  // pass 2: exp-weights + denominator
  float sum = 0.f;
  for (int i = lane; i < deg; i += 32) {
    float e = f1h + f2[h * Nn + s_nbr[i]];
    e = (e >= 0.f) ? e : LRELU_ALPHA * e;
    const float w = __expf(e - mx);
    s_w[h * CAP + i] = w;
    sum += w;
  }
#pragma unroll
  for (int o = 16; o > 0; o >>= 1) sum += __shfl_xor(sum, o, 32);
  __syncthreads();

  // pass 3: weighted aggregation; lanes 0..15 each own one output dim d
  if (lane < Dd) {
    float acc = 0.f;
    for (int i = 0; i < deg; ++i) {
      acc += s_w[h * CAP + i] * Wh1[((size_t)h * Nn + s_nbr[i]) * Dd + lane];
    }
    float o = acc / sum;
    o = (o > 0.f) ? o : (__expf(o) - 1.f);  // ELU
    hcat[(size_t)n * (Hh * Dd) + h * Dd + lane] = o;
  }
}

// ---------------------------------------------------------------------------
// Layer-2 sparse attention + log_softmax.  One wave per node.
// ---------------------------------------------------------------------------
__global__ __launch_bounds__(32) void gat_attn2(const int* __restrict__ cnt,
                                                const int* __restrict__ nbrs,
                                                const float* __restrict__ f1o,
                                                const float* __restrict__ f2o,
                                                const float* __restrict__ Wh2,
                                                float* __restrict__ out) {
  __shared__ int   s_nbr[CAP];
  __shared__ float s_w[CAP];

  const int n    = blockIdx.x;
  const int lane = threadIdx.x;
  const int deg  = min(cnt[n], CAP);
  for (int i = lane; i < deg; i += 32) s_nbr[i] = nbrs[n * CAP + i];
  __syncthreads();

  const float f1h = f1o[n];
  float mx = -3.4e38f;
  for (int i = lane; i < deg; i += 32) {
    float e = f1h + f2o[s_nbr[i]];
    e = (e >= 0.f) ? e : LRELU_ALPHA * e;
    mx = fmaxf(mx, e);
  }
#pragma unroll
  for (int o = 16; o > 0; o >>= 1) mx = fmaxf(mx, __shfl_xor(mx, o, 32));

  float sum = 0.f;
  for (int i = lane; i < deg; i += 32) {
    float e = f1h + f2o[s_nbr[i]];
    e = (e >= 0.f) ? e : LRELU_ALPHA * e;
    const float w = __expf(e - mx);
    s_w[i] = w;
    sum += w;
  }
#pragma unroll
  for (int o = 16; o > 0; o >>= 1) sum += __shfl_xor(sum, o, 32);
  __syncthreads();

  float v = -3.4e38f;
  if (lane < Cc) {
    float acc = 0.f;
    for (int i = 0; i < deg; ++i) acc += s_w[i] * Wh2[(size_t)s_nbr[i] * Cc + lane];
    v = acc / sum;
  }
  // log_softmax across the 16 class lanes
  float m2 = v;
#pragma unroll
  for (int o = 16; o > 0; o >>= 1) m2 = fmaxf(m2, __shfl_xor(m2, o, 32));
  float ex = (lane < Cc) ? __expf(v - m2) : 0.f;
  float s2 = ex;
#pragma unroll
  for (int o = 16; o > 0; o >>= 1) s2 += __shfl_xor(s2, o, 32);
  if (lane < Cc) out[(size_t)n * Cc + lane] = (v - m2) - __logf(s2);
}

// ---------------------------------------------------------------------------
extern "C" void kernel_launch(void* const* d_in, const int* in_sizes, int n_in,
                              void* d_out, int out_size, void* d_ws, size_t ws_size,
                              hipStream_t stream) {
  const float* x     = (const float*)d_in[0];
  const int*   adj   = (const int*)d_in[1];
  const float* W1    = (const float*)d_in[2];
  const float* a1    = (const float*)d_in[3];
  const float* Wout  = (const float*)d_in[4];
  const float* a_out = (const float*)d_in[5];
  float* out = (float*)d_out;

  char* p = (char*)d_ws;
  int*   cnt  = (int*)p;   p += (size_t)Nn * sizeof(int);
  int*   nbrs = (int*)p;   p += (size_t)Nn * CAP * sizeof(int);
  float* Wh1  = (float*)p; p += (size_t)Hh * Nn * Dd * sizeof(float);
  float* f1   = (float*)p; p += (size_t)Hh * Nn * sizeof(float);
  float* f2   = (float*)p; p += (size_t)Hh * Nn * sizeof(float);
  float* hcat = (float*)p; p += (size_t)Nn * Hh * Dd * sizeof(float);
  float* Wh2  = (float*)p; p += (size_t)Nn * Cc * sizeof(float);
  float* f1o  = (float*)p; p += (size_t)Nn * sizeof(float);
  float* f2o  = (float*)p; p += (size_t)Nn * sizeof(float);

  hipMemsetAsync(cnt, 0, (size_t)Nn * sizeof(int), stream);
  gat_csr<<<Nn, 256, 0, stream>>>(adj, cnt, nbrs);
  gat_gemm1<<<dim3(Nn / 16, Hh), 32, 0, stream>>>(x, W1, Wh1);
  gat_scores1<<<(Hh * Nn + 255) / 256, 256, 0, stream>>>(Wh1, a1, f1, f2);
  gat_attn1<<<Nn, 256, 0, stream>>>(cnt, nbrs, f1, f2, Wh1, hcat);
  gat_gemm2<<<Nn / 16, 32, 0, stream>>>(hcat, Wout, Wh2);
  gat_scores2<<<(Nn + 255) / 256, 256, 0, stream>>>(Wh2, a_out, f1o, f2o);
  gat_attn2<<<Nn, 32, 0, stream>>>(cnt, nbrs, f1o, f2o, Wh2, out);
}